// CombinedLoss_47545287967152
// MI455X (gfx1250) — compile-verified
//
#include <hip/hip_runtime.h>
#include <hip/hip_bf16.h>

// ---------------------------------------------------------------------------
// CombinedLoss (dice + boundary(level-set EDT) + BCE) for B=16, C=1, H=W=256
// ---------------------------------------------------------------------------
typedef __attribute__((ext_vector_type(2))) float v2f;
typedef __attribute__((ext_vector_type(8))) float v8f;

#define BATCH   16
#define HDIM    256
#define WDIM    256
#define NPIX    (BATCH * HDIM * WDIM)   // 1,048,576
#define NB3     1024                    // blocks for loss-partials kernel
#define BIGF    512.0f                  // H + W, matches reference

// Deterministic 32-lane sum using the CDNA5 WMMA unit.
// A (16x4 f32, 2 VGPRs): vgpr0 = {lanes0-15 -> A[m,0], lanes16-31 -> A[m,2]},
// vgpr1 = 0. B = ones => D[m,n] = s[m] + s[m+16].
// Per-lane sum of the 8 C/D VGPRs gives column sums over m=0..7 (lanes 0-15)
// and m=8..15 (lanes 16-31); one xor-16 shuffle completes the reduction.
__device__ __forceinline__ float wave_sum32(float s) {
    v2f a; a[0] = s;    a[1] = 0.0f;
    v2f b; b[0] = 1.0f; b[1] = 1.0f;
    v8f c = {};
    c = __builtin_amdgcn_wmma_f32_16x16x4_f32(
            /*neg_a=*/false, a, /*neg_b=*/false, b,
            /*c_mod=*/(short)0, c, /*reuse_a=*/false, /*reuse_b=*/false);
    float t = ((c[0] + c[1]) + (c[2] + c[3])) + ((c[4] + c[5]) + (c[6] + c[7]));
    t += __shfl_xor(t, 16, 32);
    return t;  // full 32-lane sum in every lane
}

// ---------------------------------------------------------------------------
// K1: per-column 1D distance (both directions), for mask and inverse mask.
// grid = BATCH blocks, block = WDIM threads (thread = one column of one image)
// Writes d1sq (squared 1D column distance). Also anymask[b], maxabs[b]=0.
// ---------------------------------------------------------------------------
__global__ void __launch_bounds__(WDIM)
k_colpass(const float* __restrict__ targ, float* __restrict__ d1in,
          float* __restrict__ d1out, unsigned* __restrict__ maxabs,
          int* __restrict__ anymask) {
    const int b = blockIdx.x;
    const int c = threadIdx.x;
    const int base = b * (HDIM * WDIM) + c;

    // top-down: distance to nearest zero (of mask / of inverse mask) above
    float up_in = -BIGF, up_out = -BIGF;
    int any = 0;
    for (int i = 0; i < HDIM; ++i) {
        float m = targ[base + i * WDIM];
        if (m == 0.0f) up_in = (float)i; else { up_out = (float)i; any = 1; }
        d1in [base + i * WDIM] = (float)i - up_in;    // >= BIG if none above
        d1out[base + i * WDIM] = (float)i - up_out;
    }
    // bottom-up: min with distance to nearest zero below, clamp to BIG, square
    float dn_in = 2.0f * BIGF, dn_out = 2.0f * BIGF;
    for (int i = HDIM - 1; i >= 0; --i) {
        float m = targ[base + i * WDIM];
        if (m == 0.0f) dn_in = (float)i; else dn_out = (float)i;
        float di = fminf(fminf(d1in [base + i * WDIM], dn_in  - (float)i), BIGF);
        float dq = fminf(fminf(d1out[base + i * WDIM], dn_out - (float)i), BIGF);
        d1in [base + i * WDIM] = di * di;
        d1out[base + i * WDIM] = dq * dq;
    }

    __shared__ int sany[WDIM];
    sany[c] = any;
    __syncthreads();
    for (int off = WDIM / 2; off > 0; off >>= 1) {
        if (c < off) sany[c] |= sany[c + off];
        __syncthreads();
    }
    if (c == 0) { anymask[b] = sany[0]; maxabs[b] = 0u; }
}

// ---------------------------------------------------------------------------
// K2: row-wise min-plus pass (Felzenszwalb identity) + phi + per-image max|phi|
// grid = BATCH*HDIM blocks (one per row), block = WDIM threads (thread = j)
// phi is written in place over d1in (row staged through LDS first -> safe).
// ---------------------------------------------------------------------------
__global__ void __launch_bounds__(WDIM)
k_rowdt(float* __restrict__ d1in, const float* __restrict__ d1out,
        unsigned* __restrict__ maxabs) {
    const int row  = blockIdx.x;          // b*HDIM + i
    const int b    = row >> 8;
    const int base = row * WDIM;
    const int j    = threadIdx.x;

    __shared__ float s_in[WDIM];
    __shared__ float s_out[WDIM];
    s_in [j] = d1in [base + j];
    s_out[j] = d1out[base + j];
    __syncthreads();

    const float jf = (float)j;
    float mi = 3.4e38f, mo = 3.4e38f;
#pragma unroll 8
    for (int k = 0; k < WDIM; ++k) {
        float d  = jf - (float)k;
        float dd = d * d;                 // same k across lanes -> LDS broadcast
        mi = fminf(mi, s_in [k] + dd);
        mo = fminf(mo, s_out[k] + dd);
    }
    // phi = dist_outside - dist_inside
    float phi = sqrtf(mo) - sqrtf(mi);
    d1in[base + j] = phi;

    __shared__ float smax[WDIM];
    smax[j] = fabsf(phi);
    __syncthreads();
    for (int off = WDIM / 2; off > 0; off >>= 1) {
        if (j < off) smax[j] = fmaxf(smax[j], smax[j + off]);
        __syncthreads();
    }
    // |phi| >= 0: IEEE bits of non-negative floats order like unsigned ints
    if (j == 0) atomicMax(&maxabs[b], __float_as_uint(smax[0]));
}

// ---------------------------------------------------------------------------
// K3: elementwise loss terms + deterministic block partial sums (WMMA reduce)
// grid = NB3 blocks x 256 threads; N/(NB3*256) == 4 exactly -> no divergence.
// ---------------------------------------------------------------------------
__global__ void __launch_bounds__(256)
k_loss_partials(const float* __restrict__ pred, const float* __restrict__ targ,
                const float* __restrict__ phi, const unsigned* __restrict__ maxabs,
                const int* __restrict__ anymask, float* __restrict__ partials) {
    const int tid    = threadIdx.x;
    const int stride = NB3 * 256;
    int idx = blockIdx.x * 256 + tid;

    float s0 = 0.f, s1 = 0.f, s2 = 0.f, s3 = 0.f, s4 = 0.f;
#pragma unroll
    for (int it = 0; it < NPIX / (NB3 * 256); ++it, idx += stride) {
        float x = pred[idx];
        float t = targ[idx];
        float p = 1.0f / (1.0f + __expf(-x));       // sigmoid
        int   bimg = idx >> 16;                     // idx / (H*W)
        float ma   = __uint_as_float(maxabs[bimg]);
        float ph   = anymask[bimg] ? (phi[idx] / (ma + 1e-8f)) : 0.0f;
        s0 += p * t;                                // intersection
        s1 += p;                                    // sum probs
        s2 += t;                                    // sum targets
        s3 += ph * p;                               // boundary numerator
        s4 += fmaxf(x, 0.0f) - x * t + log1pf(__expf(-fabsf(x)));  // BCE
    }

    // wave-level WMMA reductions (EXEC all ones: full block, no divergence)
    float r0 = wave_sum32(s0);
    float r1 = wave_sum32(s1);
    float r2 = wave_sum32(s2);
    float r3 = wave_sum32(s3);
    float r4 = wave_sum32(s4);

    __shared__ float wsum[5][8];
    const int wave = tid >> 5, lane = tid & 31;
    if (lane == 0) {
        wsum[0][wave] = r0; wsum[1][wave] = r1; wsum[2][wave] = r2;
        wsum[3][wave] = r3; wsum[4][wave] = r4;
    }
    __syncthreads();
    if (tid == 0) {
#pragma unroll
        for (int l = 0; l < 5; ++l) {
            float a = 0.f;
#pragma unroll
            for (int w = 0; w < 8; ++w) a += wsum[l][w];
            partials[blockIdx.x * 5 + l] = a;       // deterministic (no atomics)
        }
    }
}

// ---------------------------------------------------------------------------
// K4: final reduction of NB3 partials and loss combination.
// ---------------------------------------------------------------------------
__global__ void __launch_bounds__(256)
k_final(const float* __restrict__ partials, float* __restrict__ out) {
    const int tid = threadIdx.x;
    float acc[5] = {0.f, 0.f, 0.f, 0.f, 0.f};
    for (int i = tid; i < NB3; i += 256)
#pragma unroll
        for (int l = 0; l < 5; ++l) acc[l] += partials[i * 5 + l];

    __shared__ float wsum[5][8];
    const int wave = tid >> 5, lane = tid & 31;
#pragma unroll
    for (int l = 0; l < 5; ++l) {
        float r = wave_sum32(acc[l]);
        if (lane == 0) wsum[l][wave] = r;
    }
    __syncthreads();
    if (tid == 0) {
        float S[5];
#pragma unroll
        for (int l = 0; l < 5; ++l) {
            float a = 0.f;
#pragma unroll
            for (int w = 0; w < 8; ++w) a += wsum[l][w];
            S[l] = a;
        }
        const float Nf    = (float)NPIX;
        const float smooth = 1e-6f;
        float dice     = 1.0f - (2.0f * S[0] + smooth) / (S[1] + S[2] + smooth);
        float boundary = S[3] / Nf;
        float bce      = S[4] / Nf;
        // alpha = 0.005, beta = 0.6 - alpha = 0.595, 1-beta = 0.405
        out[0] = 0.405f * dice + 0.595f * boundary + 0.4f * bce;
    }
}

// ---------------------------------------------------------------------------
extern "C" void kernel_launch(void* const* d_in, const int* in_sizes, int n_in,
                              void* d_out, int out_size, void* d_ws, size_t ws_size,
                              hipStream_t stream) {
    (void)in_sizes; (void)n_in; (void)out_size; (void)ws_size;
    const float* pred = (const float*)d_in[0];
    const float* targ = (const float*)d_in[1];
    float*       out  = (float*)d_out;

    char* ws = (char*)d_ws;
    float*    d1in     = (float*)(ws);                                  // 4 MB (phi reuses this)
    float*    d1out    = (float*)(ws + (size_t)NPIX * 4);               // 4 MB
    unsigned* maxabs   = (unsigned*)(ws + (size_t)NPIX * 8);            // 16 u32
    int*      anymask  = (int*)     (ws + (size_t)NPIX * 8 + 64);       // 16 i32
    float*    partials = (float*)   (ws + (size_t)NPIX * 8 + 128);      // NB3*5 f32

    k_colpass      <<<BATCH,        WDIM, 0, stream>>>(targ, d1in, d1out, maxabs, anymask);
    k_rowdt        <<<BATCH * HDIM, WDIM, 0, stream>>>(d1in, d1out, maxabs);
    k_loss_partials<<<NB3,          256,  0, stream>>>(pred, targ, d1in, maxabs, anymask, partials);
    k_final        <<<1,            256,  0, stream>>>(partials, out);
}